// Net2_19791209300082
// MI455X (gfx1250) — compile-verified
//
#include <hip/hip_runtime.h>
#include <math.h>

// ---------------------------------------------------------------------------
// TransformerConv GNN (3 layers) for gfx1250 / MI455X.
//  - Node & skip projections:  fp32 WMMA GEMM (v_wmma_f32_16x16x4_f32),
//    wave-uniform fast path for full tiles (no divergent guards).
//  - Edge embedding e = edge_attr @ We recomputed on the fly (We <= 6.9 KB,
//    stays in cache) to avoid a 384 MB intermediate -> saves HBM traffic.
//  - Scatter softmax: order-preserving u32 map + atomicMax, f32 atomicAdd.
// ---------------------------------------------------------------------------

typedef float v2f_ __attribute__((ext_vector_type(2)));
typedef float v8f_ __attribute__((ext_vector_type(8)));

#define NNODES 50000
#define NEDGES 500000

// ---------------------------------------------------------------------------
// WMMA fp32 GEMM:  Y[nrows, ncols] = X[nrows, kdim] @ W[kdim, ncols] + bias
// One wave (32 lanes) computes a 16-row x 64-col tile (4 WMMA col-tiles).
// A 16x4 fp32 fragment: lane<16 -> M=lane, K=kb+0/1 in v0/v1 (kb=2*(lane>>4)).
// B 4x16 fp32 fragment: n=lane&15; lane<16 -> K=kb+0/1.
// C/D 16x16 f32: vgpr j, lanes 0-15 -> M=j, lanes 16-31 -> M=8+j.
// ---------------------------------------------------------------------------
__global__ __launch_bounds__(32) void gemm_wmma_f32(
    const float* __restrict__ X, const float* __restrict__ W,
    const float* __restrict__ bias, float* __restrict__ Y,
    int nrows, int kdim, int ncols)
{
  const int lane = threadIdx.x & 31;
  const int rowTile = blockIdx.x;          // 16 rows
  const int colTileBase = blockIdx.y * 4;  // 4 x 16 cols
  const int col0 = colTileBase * 16;

  const int m    = lane & 15;
  const int half = lane >> 4;              // 0 or 1
  const int kb   = half * 2;               // K sub-offset within a K=4 step
  const int row  = rowTile * 16 + m;

  v8f_ acc[4];
  acc[0] = (v8f_)0.0f; acc[1] = (v8f_)0.0f;
  acc[2] = (v8f_)0.0f; acc[3] = (v8f_)0.0f;

  const int  kFull    = kdim >> 2;                       // full K=4 steps
  const bool kRem     = (kdim & 3) != 0;
  const bool fullTile = (rowTile * 16 + 16 <= nrows) && (col0 + 64 <= ncols);

  if (fullTile) {
    // ---------------- wave-uniform fast path: no per-element guards -------
    const float* Xr = X + (size_t)row * kdim + kb;
    const float* Wc = W + col0 + m;
    for (int kt = 0; kt < kFull; ++kt) {
      v2f_ afrag;
      afrag.x = Xr[0];
      afrag.y = Xr[1];
      Xr += 4;
      const float* Wr = Wc + (size_t)(kt * 4 + kb) * ncols;
#pragma unroll
      for (int t = 0; t < 4; ++t) {
        v2f_ bfrag;
        bfrag.x = Wr[t * 16];
        bfrag.y = Wr[(size_t)ncols + t * 16];
        acc[t] = __builtin_amdgcn_wmma_f32_16x16x4_f32(
            false, afrag, false, bfrag, (short)0, acc[t], false, false);
      }
    }
    if (kRem) {  // last partial K-step (kdim=13 case)
      const int k0 = kFull * 4 + kb;
      v2f_ afrag;
      afrag.x = ((k0 + 0) < kdim) ? X[(size_t)row * kdim + k0 + 0] : 0.0f;
      afrag.y = ((k0 + 1) < kdim) ? X[(size_t)row * kdim + k0 + 1] : 0.0f;
#pragma unroll
      for (int t = 0; t < 4; ++t) {
        const int col = col0 + t * 16 + m;
        v2f_ bfrag;
        bfrag.x = ((k0 + 0) < kdim) ? W[(size_t)(k0 + 0) * ncols + col] : 0.0f;
        bfrag.y = ((k0 + 1) < kdim) ? W[(size_t)(k0 + 1) * ncols + col] : 0.0f;
        acc[t] = __builtin_amdgcn_wmma_f32_16x16x4_f32(
            false, afrag, false, bfrag, (short)0, acc[t], false, false);
      }
    }
    // bias hoisted out of store loop
    float bv[4];
#pragma unroll
    for (int t = 0; t < 4; ++t) bv[t] = bias ? bias[col0 + t * 16 + m] : 0.0f;
#pragma unroll
    for (int j = 0; j < 8; ++j) {
      const int r = rowTile * 16 + half * 8 + j;
#pragma unroll
      for (int t = 0; t < 4; ++t)
        Y[(size_t)r * ncols + col0 + t * 16 + m] = acc[t][j] + bv[t];
    }
  } else {
    // ---------------- guarded edge path (ncols=10 GEMMs etc.) -------------
    const int kTiles = kFull + (kRem ? 1 : 0);
    for (int kt = 0; kt < kTiles; ++kt) {
      const int k0 = kt * 4 + kb;
      v2f_ afrag;
      afrag.x = (row < nrows && (k0 + 0) < kdim) ? X[(size_t)row * kdim + k0 + 0] : 0.0f;
      afrag.y = (row < nrows && (k0 + 1) < kdim) ? X[(size_t)row * kdim + k0 + 1] : 0.0f;
#pragma unroll
      for (int t = 0; t < 4; ++t) {
        const int col = col0 + t * 16 + m;
        v2f_ bfrag;
        bfrag.x = ((k0 + 0) < kdim && col < ncols) ? W[(size_t)(k0 + 0) * ncols + col] : 0.0f;
        bfrag.y = ((k0 + 1) < kdim && col < ncols) ? W[(size_t)(k0 + 1) * ncols + col] : 0.0f;
        acc[t] = __builtin_amdgcn_wmma_f32_16x16x4_f32(
            false, afrag, false, bfrag, (short)0, acc[t], false, false);
      }
    }
    float bv[4];
#pragma unroll
    for (int t = 0; t < 4; ++t) {
      const int col = col0 + t * 16 + m;
      bv[t] = (bias && col < ncols) ? bias[col] : 0.0f;
    }
#pragma unroll
    for (int j = 0; j < 8; ++j) {
      const int r = rowTile * 16 + half * 8 + j;
      if (r < nrows) {
#pragma unroll
        for (int t = 0; t < 4; ++t) {
          const int col = col0 + t * 16 + m;
          if (col < ncols) Y[(size_t)r * ncols + col] = acc[t][j] + bv[t];
        }
      }
    }
  }
}

// ---------------------------------------------------------------------------
// Utility: zero buffer
// ---------------------------------------------------------------------------
__global__ void zero_f(float* __restrict__ p, long n)
{
  long i = (long)blockIdx.x * blockDim.x + threadIdx.x;
  if (i < n) p[i] = 0.0f;
}

// Order-preserving float <-> u32 mapping for atomicMax-based segment max.
__device__ __forceinline__ unsigned fmap_u(float f)
{
  unsigned u = __float_as_uint(f);
  return (u & 0x80000000u) ? ~u : (u | 0x80000000u);
}
__device__ __forceinline__ float funmap_u(unsigned u)
{
  return (u & 0x80000000u) ? __uint_as_float(u & 0x7FFFFFFFu)
                           : __uint_as_float(~u);
}

// ---------------------------------------------------------------------------
// Per-edge raw attention score:
//   alpha[e,h] = scale * dot(q[dst], k[src] + edge_attr[e] @ We[:, h*C:..])
// fused with segment max (atomicMax on mapped u32).
// ---------------------------------------------------------------------------
__global__ void edge_dot(const float* __restrict__ q, const float* __restrict__ k,
                         const float* __restrict__ ea, const float* __restrict__ We,
                         const int* __restrict__ src, const int* __restrict__ dst,
                         float* __restrict__ alpha, unsigned* __restrict__ amax_u,
                         int E, int H, int C, float scale)
{
  int idx = blockIdx.x * blockDim.x + threadIdx.x;  // e*H + h
  if (idx >= E * H) return;
  const int e = idx / H;
  const int h = idx - e * H;
  const int HC = H * C;
  const int s = src[e];
  const int d = dst[e];

  float eav[9];
#pragma unroll
  for (int j = 0; j < 9; ++j) eav[j] = ea[(size_t)e * 9 + j];

  const float* qp  = q + (size_t)d * HC + h * C;
  const float* kp  = k + (size_t)s * HC + h * C;
  const float* Wec = We + h * C;  // row stride HC

  float acc = 0.0f;
  for (int c = 0; c < C; ++c) {
    float ev = 0.0f;
#pragma unroll
    for (int j = 0; j < 9; ++j) ev = fmaf(eav[j], Wec[(size_t)j * HC + c], ev);
    acc = fmaf(qp[c], kp[c] + ev, acc);
  }
  acc *= scale;
  alpha[idx] = acc;
  atomicMax(&amax_u[(size_t)d * H + h], fmap_u(acc));
}

// Convert mapped-u32 max back to float; non-finite (incl. no-edge nodes) -> 0.
__global__ void amax_fix(unsigned* __restrict__ amax, long n)
{
  long i = (long)blockIdx.x * blockDim.x + threadIdx.x;
  if (i >= n) return;
  float f = funmap_u(amax[i]);
  if (!__builtin_isfinite(f)) f = 0.0f;
  ((float*)amax)[i] = f;
}

// ex = exp(alpha - amax[dst]); alpha <- ex; denom[dst] += ex
__global__ void edge_exp(float* __restrict__ alpha, const float* __restrict__ amaxf,
                         float* __restrict__ denom, const int* __restrict__ dst,
                         int E, int H)
{
  int idx = blockIdx.x * blockDim.x + threadIdx.x;
  if (idx >= E * H) return;
  const int e = idx / H;
  const int h = idx - e * H;
  const int d = dst[e];
  float x = __expf(alpha[idx] - amaxf[(size_t)d * H + h]);
  alpha[idx] = x;
  atomicAdd(&denom[(size_t)d * H + h], x);
}

// out[dst, i] += (v[src, i] + e[e, i]) * ex[e,h] / (denom[dst,h] + 1e-16)
__global__ void edge_message(const float* __restrict__ v, const float* __restrict__ ea,
                             const float* __restrict__ We,
                             const int* __restrict__ src, const int* __restrict__ dst,
                             const float* __restrict__ ex, const float* __restrict__ denom,
                             float* __restrict__ out, int E, int H, int C)
{
  const int HC = H * C;
  long idx = (long)blockIdx.x * blockDim.x + threadIdx.x;  // e*HC + i
  if (idx >= (long)E * HC) return;
  const int e = (int)(idx / HC);
  const int i = (int)(idx - (long)e * HC);
  const int h = i / C;
  const int s = src[e];
  const int d = dst[e];

  float ev = 0.0f;
#pragma unroll
  for (int j = 0; j < 9; ++j)
    ev = fmaf(ea[(size_t)e * 9 + j], We[(size_t)j * HC + i], ev);

  const float a = ex[(size_t)e * H + h] / (denom[(size_t)d * H + h] + 1e-16f);
  const float msg = (v[(size_t)s * HC + i] + ev) * a;
  atomicAdd(&out[(size_t)d * HC + i], msg);
}

// Layers 1/2 combine: out = relu(mean_heads(attn) + skip) [+ resid]
__global__ void combine_mean3(const float* __restrict__ attn, const float* __restrict__ skip,
                              const float* __restrict__ resid, float* __restrict__ out,
                              int N, int C)
{
  int idx = blockIdx.x * blockDim.x + threadIdx.x;
  if (idx >= N * C) return;
  const int n = idx / C;
  const int c = idx - n * C;
  const float* a = attn + (size_t)n * 3 * C;
  float m = (a[c] + a[C + c] + a[2 * C + c]) * (1.0f / 3.0f);
  float val = fmaxf(m + skip[idx], 0.0f);
  if (resid) val += resid[idx];
  out[idx] = val;
}

// Layer 3 combine + log_softmax over 10 classes
__global__ void final_logsoftmax(const float* __restrict__ attn, const float* __restrict__ skip,
                                 float* __restrict__ out, int N)
{
  int n = blockIdx.x * blockDim.x + threadIdx.x;
  if (n >= N) return;
  float vals[10];
  float mx = -3.402823466e+38f;
#pragma unroll
  for (int i = 0; i < 10; ++i) {
    vals[i] = attn[(size_t)n * 10 + i] + skip[(size_t)n * 10 + i];
    mx = fmaxf(mx, vals[i]);
  }
  float s = 0.0f;
#pragma unroll
  for (int i = 0; i < 10; ++i) s += __expf(vals[i] - mx);
  const float lse = mx + __logf(s);
#pragma unroll
  for (int i = 0; i < 10; ++i) out[(size_t)n * 10 + i] = vals[i] - lse;
}

// ---------------------------------------------------------------------------
// Host-side helpers
// ---------------------------------------------------------------------------
static void launch_gemm(const float* X, const float* W, const float* b, float* Y,
                        int nrows, int kdim, int ncols, hipStream_t s)
{
  dim3 grid((nrows + 15) / 16, (ncols + 63) / 64);
  gemm_wmma_f32<<<grid, 32, 0, s>>>(X, W, b, Y, nrows, kdim, ncols);
}

static void run_edge_phase(const float* q, const float* k, const float* v,
                           const float* ea, const float* We,
                           const int* src, const int* dst,
                           float* alpha, float* amaxb, float* denom, float* attn,
                           int N, int E, int H, int C, hipStream_t s)
{
  const int HC = H * C;
  const long nAttn = (long)N * HC;
  const long nNH = (long)N * H;
  zero_f<<<(unsigned)((nAttn + 255) / 256), 256, 0, s>>>(attn, nAttn);
  zero_f<<<(unsigned)((nNH + 255) / 256), 256, 0, s>>>(amaxb, nNH);
  zero_f<<<(unsigned)((nNH + 255) / 256), 256, 0, s>>>(denom, nNH);

  const int EH = E * H;
  const float scale = 1.0f / sqrtf((float)C);
  edge_dot<<<(EH + 255) / 256, 256, 0, s>>>(q, k, ea, We, src, dst, alpha,
                                            (unsigned*)amaxb, E, H, C, scale);
  amax_fix<<<(unsigned)((nNH + 255) / 256), 256, 0, s>>>((unsigned*)amaxb, nNH);
  edge_exp<<<(EH + 255) / 256, 256, 0, s>>>(alpha, amaxb, denom, dst, E, H);

  const long tot = (long)E * HC;
  edge_message<<<(unsigned)((tot + 255) / 256), 256, 0, s>>>(v, ea, We, src, dst,
                                                             alpha, denom, attn, E, H, C);
}

extern "C" void kernel_launch(void* const* d_in, const int* in_sizes, int n_in,
                              void* d_out, int out_size, void* d_ws, size_t ws_size,
                              hipStream_t stream)
{
  (void)in_sizes; (void)n_in; (void)out_size; (void)ws_size;
  const int N = NNODES, E = NEDGES;

  const float* x   = (const float*)d_in[0];
  const int*   ei  = (const int*)d_in[1];
  const int*   src = ei;
  const int*   dst = ei + E;
  const float* ea  = (const float*)d_in[2];

  const float* Wq1 = (const float*)d_in[3];  const float* bq1 = (const float*)d_in[4];
  const float* Wk1 = (const float*)d_in[5];  const float* bk1 = (const float*)d_in[6];
  const float* Wv1 = (const float*)d_in[7];  const float* bv1 = (const float*)d_in[8];
  const float* We1 = (const float*)d_in[9];
  const float* Ws1 = (const float*)d_in[10]; const float* bs1 = (const float*)d_in[11];

  const float* Wq2 = (const float*)d_in[12]; const float* bq2 = (const float*)d_in[13];
  const float* Wk2 = (const float*)d_in[14]; const float* bk2 = (const float*)d_in[15];
  const float* Wv2 = (const float*)d_in[16]; const float* bv2 = (const float*)d_in[17];
  const float* We2 = (const float*)d_in[18];
  const float* Ws2 = (const float*)d_in[19]; const float* bs2 = (const float*)d_in[20];

  const float* Wq3 = (const float*)d_in[21]; const float* bq3 = (const float*)d_in[22];
  const float* Wk3 = (const float*)d_in[23]; const float* bk3 = (const float*)d_in[24];
  const float* Wv3 = (const float*)d_in[25]; const float* bv3 = (const float*)d_in[26];
  const float* We3 = (const float*)d_in[27];
  const float* Ws3 = (const float*)d_in[28]; const float* bs3 = (const float*)d_in[29];

  // Workspace layout (floats): ~200 MB total
  float* ws    = (float*)d_ws;
  float* q     = ws;                         // N*192
  float* k     = q     + (size_t)N * 192;    // N*192
  float* v     = k     + (size_t)N * 192;    // N*192
  float* attn  = v     + (size_t)N * 192;    // N*192
  float* alpha = attn  + (size_t)N * 192;    // E*3
  float* amax  = alpha + (size_t)E * 3;      // N*3 (u32-aliased)
  float* denom = amax  + (size_t)N * 3;      // N*3
  float* h1b   = denom + (size_t)N * 3;      // N*64
  float* hsum  = h1b   + (size_t)N * 64;     // N*64
  float* skipb = hsum  + (size_t)N * 64;     // N*64

  // ---------------- Layer 1: in=13, H=3, C=64, concat=False ----------------
  launch_gemm(x, Wq1, bq1, q, N, 13, 192, stream);
  launch_gemm(x, Wk1, bk1, k, N, 13, 192, stream);
  launch_gemm(x, Wv1, bv1, v, N, 13, 192, stream);
  launch_gemm(x, Ws1, bs1, skipb, N, 13, 64, stream);
  run_edge_phase(q, k, v, ea, We1, src, dst, alpha, amax, denom, attn,
                 N, E, 3, 64, stream);
  combine_mean3<<<(N * 64 + 255) / 256, 256, 0, stream>>>(attn, skipb, nullptr,
                                                          h1b, N, 64);

  // ---------------- Layer 2: in=64, H=3, C=64, concat=False ----------------
  launch_gemm(h1b, Wq2, bq2, q, N, 64, 192, stream);
  launch_gemm(h1b, Wk2, bk2, k, N, 64, 192, stream);
  launch_gemm(h1b, Wv2, bv2, v, N, 64, 192, stream);
  launch_gemm(h1b, Ws2, bs2, skipb, N, 64, 64, stream);
  run_edge_phase(q, k, v, ea, We2, src, dst, alpha, amax, denom, attn,
                 N, E, 3, 64, stream);
  // h = relu(conv2(h1)) + h1
  combine_mean3<<<(N * 64 + 255) / 256, 256, 0, stream>>>(attn, skipb, h1b,
                                                          hsum, N, 64);

  // ---------------- Layer 3: in=64, H=1, C=10, concat=True -----------------
  launch_gemm(hsum, Wq3, bq3, q, N, 64, 10, stream);
  launch_gemm(hsum, Wk3, bk3, k, N, 64, 10, stream);
  launch_gemm(hsum, Wv3, bv3, v, N, 64, 10, stream);
  launch_gemm(hsum, Ws3, bs3, skipb, N, 64, 10, stream);
  run_edge_phase(q, k, v, ea, We3, src, dst, alpha, amax, denom, attn,
                 N, E, 1, 10, stream);
  final_logsoftmax<<<(N + 255) / 256, 256, 0, stream>>>(attn, skipb,
                                                        (float*)d_out, N);
}